// MeanAggregator_28518582846055
// MI455X (gfx1250) — compile-verified
//
#include <hip/hip_runtime.h>
#include <stdint.h>

namespace {
constexpr int kH       = 256;  // embedding width (reference H)
constexpr int kDeg     = 32;   // neighbors per segment (reference DEG)
constexpr int kThreads = 256;  // 8 wave32s per block
}

// One block per segment. Gathers the 32 neighbor rows (32 KB) into LDS with
// CDNA5 async global->LDS b128 copies, reduces them with float4 LDS reads,
// and writes [mean | subject] (512 floats) per segment.
__global__ __launch_bounds__(kThreads)
void MeanAggregator_seg_mean_concat(const float* __restrict__ ent,      // [N_ENT, kH]
                                    const int*   __restrict__ nbr_idx,  // [S*kDeg]
                                    const int*   __restrict__ batch_id, // [S]
                                    const int*   __restrict__ s_tem,    // [B]
                                    float*       __restrict__ out)      // [S, 2*kH]
{
    __shared__ float tile[kDeg * kH];   // 32 KB gathered neighbor rows
    __shared__ float part[4 * kH];      // 4 KB partial sums (4 row-groups)
    __shared__ int   idx[kDeg];
    __shared__ int   subjIdx;

    const int s = blockIdx.x;
    const int t = threadIdx.x;

    if (t < kDeg) idx[t] = nbr_idx[s * kDeg + t];
    if (t == kDeg) subjIdx = s_tem[batch_id[s]];
    __syncthreads();

    // ---- async gather: 32 rows x 1 KB = 2048 x 16 B chunks into LDS ----
    // GVS mode: 64-bit SGPR base + per-lane 32-bit byte offset.
    const uint64_t base    = (uint64_t)(uintptr_t)ent;
    // Generic pointers to LDS carry the wave-relative LDS byte offset in the
    // low 32 bits (aperture scheme).
    const uint32_t ldsTile = (uint32_t)(uintptr_t)(&tile[0]);

    constexpr int kChunks  = (kDeg * kH * 4) / 16;      // 2048
    constexpr int kIters   = kChunks / kThreads;        // 8 outstanding per lane
#pragma unroll
    for (int it = 0; it < kIters; ++it) {
        const int c = t + it * kThreads;                // chunk id 0..2047
        const int j = c >> 6;                           // neighbor row 0..31
        const int q = c & 63;                           // 16B quad within row
        const uint32_t goff = (uint32_t)idx[j] * (uint32_t)(kH * 4) + (uint32_t)q * 16u;
        const uint32_t loff = ldsTile + (uint32_t)c * 16u;
        asm volatile("global_load_async_to_lds_b128 %0, %1, %2"
                     :: "v"(loff), "v"(goff), "s"(base)
                     : "memory");
    }
    asm volatile("s_wait_asynccnt 0" ::: "memory");
    __syncthreads();   // make all waves' async LDS writes visible

    // ---- two-phase reduction: float4 LDS reads, 4 row-groups of 8 rows ----
    {
        const int g = t >> 6;                 // row group 0..3
        const int q = t & 63;                 // column quad 0..63
        const float4* tile4 = (const float4*)tile;
        float4 a = make_float4(0.f, 0.f, 0.f, 0.f);
#pragma unroll
        for (int jj = 0; jj < kDeg / 4; ++jj) {
            const int j = g * (kDeg / 4) + jj;
            const float4 v = tile4[j * (kH / 4) + q];
            a.x += v.x; a.y += v.y; a.z += v.z; a.w += v.w;
        }
        ((float4*)&part[g * kH])[q] = a;      // part[g][4q+k] = col (4q+k) partial
    }
    __syncthreads();

    float acc = part[0 * kH + t] + part[1 * kH + t]
              + part[2 * kH + t] + part[3 * kH + t];
    acc *= (1.0f / (float)kDeg);

    // subject embedding row: L2-hot (reused across SEQ_LEN segments per batch)
    const float subj = ent[(size_t)subjIdx * kH + t];

    float* o = out + (size_t)s * (2 * kH);
    o[t]      = acc;    // mean half
    o[kH + t] = subj;   // subject half
}

extern "C" void kernel_launch(void* const* d_in, const int* in_sizes, int n_in,
                              void* d_out, int out_size, void* d_ws, size_t ws_size,
                              hipStream_t stream) {
    (void)n_in; (void)out_size; (void)d_ws; (void)ws_size;
    const float* ent      = (const float*)d_in[0];  // ent_embeds [N_ENT*H]
    const int*   nbr_idx  = (const int*)  d_in[1];  // embeds_idx [E]
    // d_in[2] = segment_ids: implied by layout (DEG contiguous per segment)
    const int*   batch_id = (const int*)  d_in[3];  // batch_id   [S]
    // d_in[4] = seq_pos: implied (out is dense in segment order)
    const int*   s_tem    = (const int*)  d_in[5];  // s_tem      [B]
    // d_in[6] = seq_len scalar (device); not needed on host

    const int S = in_sizes[3];                      // number of segments
    MeanAggregator_seg_mean_concat<<<dim3((unsigned)S), dim3(kThreads), 0, stream>>>(
        ent, nbr_idx, batch_id, s_tem, (float*)d_out);
}